// DCNv2Pooling_67697274520437
// MI455X (gfx1250) — compile-verified
//
#include <hip/hip_runtime.h>
#include <hip/hip_bf16.h>
#include <stdint.h>

// Problem constants (fixed by the reference harness)
#define SPATIAL_SCALE 0.0625f
#define P_ 7            // POOLED
#define S_ 4            // SAMPLE
#define TRANS_STD 0.1f
#define PART_ 7         // == P_, so part_h[ph] == ph, part_w[pw] == pw

#define BATCH 2
#define CHAN 256
#define HH 160
#define WW 160
#define NROI 128
#define PLANE (HH * WW)

// ---- CDNA5 async global->LDS path, feature-probed ------------------------
#ifdef __HIP_DEVICE_COMPILE__
  #if __has_builtin(__builtin_amdgcn_global_load_async_to_lds_b32)
    #define USE_ASYNC_BUILTIN 1
  #else
    #define USE_ASYNC_BUILTIN 0
  #endif
  #if __has_builtin(__builtin_amdgcn_s_wait_asynccnt)
    #define USE_WAITASYNC_BUILTIN 1
  #else
    #define USE_WAITASYNC_BUILTIN 0
  #endif
#else
  #define USE_ASYNC_BUILTIN 0
  #define USE_WAITASYNC_BUILTIN 0
#endif

// clang prints AS1 as `__device__` in diagnostics; builtin wants int* in AS1/AS3.
typedef __attribute__((address_space(1))) int g1_int;
typedef __attribute__((address_space(3))) int l3_int;

__device__ __forceinline__ void async_load_dword_to_lds(const float* gsrc, float* lds_dst) {
#if USE_ASYNC_BUILTIN
    __builtin_amdgcn_global_load_async_to_lds_b32(
        (g1_int*)(uintptr_t)gsrc,
        (l3_int*)(uint32_t)(uintptr_t)lds_dst,
        0, 0);
#else
    uint64_t ga = (uint64_t)(uintptr_t)gsrc;
    uint32_t la = (uint32_t)(uintptr_t)lds_dst;   // low 32 bits of generic LDS addr = LDS offset
    asm volatile("global_load_async_to_lds_b32 %0, %1, off"
                 :: "v"(la), "v"(ga) : "memory");
#endif
}

__device__ __forceinline__ void wait_asynccnt0() {
#if USE_WAITASYNC_BUILTIN
    __builtin_amdgcn_s_wait_asynccnt(0);
#else
    asm volatile("s_wait_asynccnt 0x0" ::: "memory");
#endif
}

// Per-sample record: 32B => exactly two ds_load_b128 per sample.
struct alignas(16) SampleRec {
    int   o00, o01, o10, o11;     // flat offsets into one channel plane (y*W + x)
    float w00, w01, w10, w11;     // bilinear weights, pre-masked by `valid`
};

__global__ __launch_bounds__(256)
void deform_roi_pool_gfx1250(const float* __restrict__ inp,
                             const float* __restrict__ rois,
                             const float* __restrict__ offset,
                             float* __restrict__ out)
{
    __shared__ float     stage[8];     // [0..4] roi row, [5] tx_raw, [6] ty_raw, [7] valid count
    __shared__ SampleRec recs[S_ * S_];

    // 3D grid: no div/mod in the prologue
    const int pw  = blockIdx.x;
    const int ph  = blockIdx.y;
    const int n   = blockIdx.z;
    const int tid = threadIdx.x;

    // ---- Stage roi row + deform offsets into LDS via CDNA5 async DMA ----
    if (tid < 7) {
        const float* gsrc;
        if (tid < 5) {
            gsrc = rois + n * 5 + tid;
        } else {
            const int comp = tid - 5;           // 0 -> tx, 1 -> ty
            gsrc = offset + (((size_t)n * 2 + comp) * PART_ + ph) * PART_ + pw;
        }
        async_load_dword_to_lds(gsrc, &stage[tid]);
    }
    wait_asynccnt0();          // issuing wave drains ASYNCcnt; no-op for others
    __syncthreads();

    // ---- 16 lanes (wave 0) compute the 16 sample records, once per block ----
    if (tid < S_ * S_) {
        const float rx1 = stage[1], ry1 = stage[2], rx2 = stage[3], ry2 = stage[4];
        // rintf == round-half-even, matching jnp.round
        const float rsw = rintf(rx1) * SPATIAL_SCALE - 0.5f;
        const float rsh = rintf(ry1) * SPATIAL_SCALE - 0.5f;
        const float rew = (rintf(rx2) + 1.0f) * SPATIAL_SCALE - 0.5f;
        const float reh = (rintf(ry2) + 1.0f) * SPATIAL_SCALE - 0.5f;
        const float rw = fmaxf(rew - rsw, 0.1f);
        const float rh = fmaxf(reh - rsh, 0.1f);
        const float bw = rw / (float)P_;
        const float bh = rh / (float)P_;
        const float sw = bw / (float)S_;
        const float sh = bh / (float)S_;
        const float tx = stage[5] * TRANS_STD;
        const float ty = stage[6] * TRANS_STD;
        const float wstart = (float)pw * bw + rsw + tx * rw;
        const float hstart = (float)ph * bh + rsh + ty * rh;

        const int iw = tid & (S_ - 1);
        const int ih = tid >> 2;
        const float x = wstart + (float)iw * sw;
        const float y = hstart + (float)ih * sh;

        const bool ok = (x >= -0.5f) && (x <= (float)WW - 0.5f) &&
                        (y >= -0.5f) && (y <= (float)HH - 0.5f);

        // wave32 ballot over the 16 active lanes -> per-bin valid count
        const unsigned long long m = __ballot(ok);
        if (tid == 0) stage[7] = (float)__popcll(m);

        const float xc = fminf(fmaxf(x, 0.0f), (float)WW - 1.0f);
        const float yc = fminf(fmaxf(y, 0.0f), (float)HH - 1.0f);
        const float x0f = floorf(xc), y0f = floorf(yc);
        const float dx = xc - x0f,   dy = yc - y0f;
        const int x0 = (int)x0f,         y0 = (int)y0f;
        const int x1 = (int)ceilf(xc),   y1 = (int)ceilf(yc);   // == x0 when dx==0
        const float vf = ok ? 1.0f : 0.0f;

        SampleRec r;
        r.o00 = y0 * WW + x0;
        r.o01 = y1 * WW + x0;     // pairs with w01 = (1-dx)*dy  (ref: v01 = inp[y1, x0])
        r.o10 = y0 * WW + x1;
        r.o11 = y1 * WW + x1;
        r.w00 = (1.0f - dx) * (1.0f - dy) * vf;
        r.w01 = (1.0f - dx) * dy * vf;
        r.w10 = dx * (1.0f - dy) * vf;
        r.w11 = dx * dy * vf;
        recs[tid] = r;
    }
    __syncthreads();

    // ---- 256 lanes = 256 channels: 64-point weighted gather-reduce ----
    const int   b     = (int)stage[0];
    const float cnt   = stage[7];
    const int   c     = tid;
    // Block-uniform base pointer (SGPR) + 32-bit per-lane offsets -> saddr-form loads
    const float* plane = inp + (size_t)b * (CHAN * (size_t)PLANE);
    const int   cbase  = c * PLANE;

    // gfx1250 prefetch of the first gather line (global_prefetch_b8)
    __builtin_prefetch(plane + cbase + recs[0].o00, 0, 0);

    float acc = 0.0f;
#pragma unroll
    for (int s = 0; s < S_ * S_; ++s) {
        const SampleRec r = recs[s];
        const float v00 = plane[cbase + r.o00];
        const float v01 = plane[cbase + r.o01];
        const float v10 = plane[cbase + r.o10];
        const float v11 = plane[cbase + r.o11];
        acc += r.w00 * v00 + r.w01 * v01 + r.w10 * v10 + r.w11 * v11;
    }

    const float result = acc / fmaxf(cnt, 1.0f);
    // streaming (write-once) output: non-temporal store
    __builtin_nontemporal_store(result,
        out + (((size_t)n * CHAN + c) * P_ + ph) * P_ + pw);
}

extern "C" void kernel_launch(void* const* d_in, const int* in_sizes, int n_in,
                              void* d_out, int out_size, void* d_ws, size_t ws_size,
                              hipStream_t stream) {
    (void)in_sizes; (void)n_in; (void)out_size; (void)d_ws; (void)ws_size;
    const float* inp    = (const float*)d_in[0];   // (2, 256, 160, 160) f32
    const float* rois   = (const float*)d_in[1];   // (128, 5) f32
    const float* offset = (const float*)d_in[2];   // (128, 2, 7, 7) f32
    float* out = (float*)d_out;                    // (128, 256, 7, 7) f32

    dim3 grid(P_, P_, NROI);                       // (pw, ph, n): 6272 bins
    deform_roi_pool_gfx1250<<<grid, 256, 0, stream>>>(inp, rois, offset, out);
}